// AGCRN_20607253087006
// MI455X (gfx1250) — compile-verified
//
#include <hip/hip_runtime.h>
#include <hip/hip_bf16.h>
#include <cstddef>

// ---------------------------------------------------------------------------
// AGCRN forward for MI455X (gfx1250, wave32, WMMA).
// Dims: B=16, T=12, N=2048, DIN=1, DOUT=64, K=3, ED=10, H=4, HORIZON=12.
//
//   A  = softmax(relu(E E^T))                (2048x2048)
//   A2 = 2*A@A - I                           (WMMA f32 GEMM, mode=1 epilogue)
//   for t in 0..11:                          (states == last_states, run once)
//     X  = [x_t | H]; Y1 = A@X; Y2 = A2@X    (WMMA f32 GEMMs)
//     z,r = sigmoid(per-node contraction)    (gw rebuilt in LDS from emb)
//     X  = [x_t | z*H] (in place); Y1,Y2 again
//     hc = tanh(...); H = r*H + (1-r)*hc; states[t] = H
//   o1 = transformer(states) at t=11 only    (fused per-token attention)
//   out = head(o1, H)                        (o2 == o3 == H_final)
//
// Workspace requirement: ~185 MB of d_ws (all f32, offsets below).
// ---------------------------------------------------------------------------

#define NB    2048          // nodes
#define BB    16            // batch
#define TT    12            // timesteps
#define CIN   65            // DIN + DOUT
#define DOUTC 64
#define XCOLS (BB*CIN)      // 1040

typedef float v2f __attribute__((ext_vector_type(2)));
typedef float v8f __attribute__((ext_vector_type(8)));

// ws offsets (in floats)
#define OFF_A   ((size_t)0)
#define OFF_A2  ((size_t)4194304)
#define OFF_X   ((size_t)8388608)
#define OFF_Y1  ((size_t)10518528)
#define OFF_Y2  ((size_t)12648448)
#define OFF_R   ((size_t)14778368)
#define OFF_H   ((size_t)16875520)
#define OFF_ST  ((size_t)18972672)
#define OFF_O1  ((size_t)44138496)

// ---------------------------------------------------------------------------
// A[n,:] = softmax(relu(E[n,:] @ E^T)) ; one block per row n
// ---------------------------------------------------------------------------
__global__ __launch_bounds__(256) void k_softmax_adj(const float* __restrict__ E,
                                                     float* __restrict__ A) {
  __shared__ float red[256];
  const int n = blockIdx.x, tid = threadIdx.x;
  float e[10];
  #pragma unroll
  for (int d = 0; d < 10; ++d) e[d] = E[n*10 + d];
  float s[8];
  float lmax = -1e30f;
  #pragma unroll
  for (int j = 0; j < 8; ++j) {
    const int m = tid + j*256;
    float dot = 0.f;
    #pragma unroll
    for (int d = 0; d < 10; ++d) dot += e[d] * E[m*10 + d];
    dot = fmaxf(dot, 0.f);
    s[j] = dot;
    lmax = fmaxf(lmax, dot);
  }
  red[tid] = lmax; __syncthreads();
  for (int off = 128; off > 0; off >>= 1) {
    if (tid < off) red[tid] = fmaxf(red[tid], red[tid+off]);
    __syncthreads();
  }
  const float mx = red[0]; __syncthreads();
  float lsum = 0.f;
  #pragma unroll
  for (int j = 0; j < 8; ++j) { s[j] = expf(s[j] - mx); lsum += s[j]; }
  red[tid] = lsum; __syncthreads();
  for (int off = 128; off > 0; off >>= 1) {
    if (tid < off) red[tid] += red[tid+off];
    __syncthreads();
  }
  const float inv = 1.f / red[0];
  #pragma unroll
  for (int j = 0; j < 8; ++j) A[(size_t)n*NB + tid + j*256] = s[j] * inv;
}

// ---------------------------------------------------------------------------
// WMMA f32 GEMM: D = Ag(2048 x K, row-major, lda=K) @ Bg(K x Ncols, ldb)
//   mode 0: D = acc
//   mode 1: D = 2*acc - I        (for A2 = 2*A@A - I)
// Block: 256 threads = 8 waves. Tile BM=256 (each wave owns TWO 16-row
// fragments -> two independent WMMA accumulator chains, shared B fragment),
// BN=16, BK=32. Grid: (M/256, Ncols/16); all dims divide exactly.
// Fragment layouts per CDNA5 ISA (wave32):
//   A 16x4 : lane l -> row = l&15, v={A[row][2*(l>>4)], A[row][2*(l>>4)+1]}
//   B 4x16 : lane l -> col = l&15, v={B[2*(l>>4)][col], B[2*(l>>4)+1][col]}
//   C 16x16: vgpr r, lane l -> D[r + 8*(l>>4)][l&15]
// ---------------------------------------------------------------------------
__global__ __launch_bounds__(256) void k_gemm_wmma(const float* __restrict__ Ag,
                                                   const float* __restrict__ Bg,
                                                   float* __restrict__ Dg,
                                                   int K, int ldb, int ldd, int mode) {
  __shared__ __align__(16) float As[256*32];   // 32 KB
  __shared__ __align__(16) float Bs[32*16];    // 2 KB
  const int tid  = threadIdx.x;
  const int wave = tid >> 5;
  const int l    = tid & 31;
  const int m0   = blockIdx.x * 256;
  const int n0   = blockIdx.y * 16;
  const int lda  = K;
  const int row  = l & 15;     // A row / B,D column within fragment
  const int kp   = l >> 4;     // lane-half selector

  v8f acc0 = {0.f,0.f,0.f,0.f,0.f,0.f,0.f,0.f};
  v8f acc1 = {0.f,0.f,0.f,0.f,0.f,0.f,0.f,0.f};

  for (int k0 = 0; k0 < K; k0 += 32) {
    // cooperative load A tile (256x32) as float4s; prefetch next k-stripe
    #pragma unroll
    for (int j = 0; j < 8; ++j) {
      const int f  = tid + j*256;       // float4 index, 0..2047
      const int r  = f >> 3;
      const int c4 = f & 7;
      const float* gp = Ag + (size_t)(m0 + r)*lda + k0 + c4*4;
      const float4 v = *reinterpret_cast<const float4*>(gp);
      *reinterpret_cast<float4*>(&As[r*32 + c4*4]) = v;
      if (k0 + 32 < K) __builtin_prefetch(gp + 32, 0, 1);   // global_prefetch_b8
    }
    // cooperative load B tile (32x16)
    if (tid < 128) {
      const int r  = tid >> 2;
      const int c4 = tid & 3;
      const float* gp = Bg + (size_t)(k0 + r)*ldb + n0 + c4*4;
      const float4 v = *reinterpret_cast<const float4*>(gp);
      *reinterpret_cast<float4*>(&Bs[r*16 + c4*4]) = v;
      if (k0 + 32 < K) __builtin_prefetch(gp + (size_t)32*ldb, 0, 1);
    }
    __syncthreads();

    const float* arow0 = &As[(wave*32 + row)*32];
    const float* arow1 = &As[(wave*32 + 16 + row)*32];
    #pragma unroll
    for (int kk = 0; kk < 8; ++kk) {
      const int kb = kk*4 + kp*2;
      v2f b, a0, a1;
      b.x  = Bs[kb*16 + row];
      b.y  = Bs[(kb + 1)*16 + row];
      a0.x = arow0[kb];  a0.y = arow0[kb + 1];
      a1.x = arow1[kb];  a1.y = arow1[kb + 1];
      acc0 = __builtin_amdgcn_wmma_f32_16x16x4_f32(
          false, a0, false, b, (short)0, acc0, false, false);
      acc1 = __builtin_amdgcn_wmma_f32_16x16x4_f32(
          false, a1, false, b, (short)0, acc1, false, false);
    }
    __syncthreads();
  }

  const int col = l & 15;
  const int gn  = n0 + col;
  #pragma unroll
  for (int r8 = 0; r8 < 8; ++r8) {
    const int gm0 = m0 + wave*32      + r8 + kp*8;
    const int gm1 = m0 + wave*32 + 16 + r8 + kp*8;
    float v0 = acc0[r8];
    float v1 = acc1[r8];
    if (mode == 1) {
      v0 = 2.f*v0 - ((gm0 == gn) ? 1.f : 0.f);
      v1 = 2.f*v1 - ((gm1 == gn) ? 1.f : 0.f);
    }
    Dg[(size_t)gm0*ldd + gn] = v0;
    Dg[(size_t)gm1*ldd + gn] = v1;
  }
}

// ---------------------------------------------------------------------------
__global__ void k_zero(float* __restrict__ p, int n) {
  const int g = blockIdx.x*256 + threadIdx.x;
  if (g < n) p[g] = 0.f;
}

// X[n, b*65 + i] = (i==0) ? source[b,t,n,0] : H[n,b,i-1]
__global__ void k_build_X(float* __restrict__ X, const float* __restrict__ src,
                          const float* __restrict__ H, int t) {
  const int gid = blockIdx.x*256 + threadIdx.x;
  if (gid >= NB*XCOLS) return;
  const int n = gid / XCOLS, rem = gid % XCOLS;
  const int b = rem / CIN,   i   = rem % CIN;
  float v;
  if (i == 0) v = src[((size_t)(b*TT + t))*NB + n];
  else        v = H[((size_t)n*BB + b)*DOUTC + (i - 1)];
  X[gid] = v;
}

// ---------------------------------------------------------------------------
// Gate: zr = sigmoid(sum_k sum_i Yk[n,b,i] * gw[n,k,i,o] + gb[n,o])
// gw/gb rebuilt in LDS from 10-dim embedding. Writes r -> Rbuf and
// overwrites X (cols 1..64 per batch) with z*H in place (col 0 = x_t kept).
// dynamic LDS: (24960 + 128 + 3*1040) floats = 112832 B
// ---------------------------------------------------------------------------
__global__ __launch_bounds__(256) void k_gate(const float* __restrict__ E,
                                              const float* __restrict__ gate_w,
                                              const float* __restrict__ gate_b,
                                              const float* __restrict__ Y1,
                                              const float* __restrict__ Y2,
                                              const float* __restrict__ H,
                                              float* __restrict__ Rbuf,
                                              float* __restrict__ X /*in+out*/) {
  extern __shared__ float sm[];
  float* sgw = sm;               // 3*65*128 = 24960
  float* sgb = sgw + 24960;      // 128
  float* sx  = sgb + 128;        // 1040
  float* sy1 = sx  + 1040;       // 1040
  float* sy2 = sy1 + 1040;       // 1040
  const int n = blockIdx.x, tid = threadIdx.x;
  float e[10];
  #pragma unroll
  for (int d = 0; d < 10; ++d) e[d] = E[n*10 + d];

  for (int idx = tid; idx < 24960; idx += 256) {
    float acc = 0.f;
    #pragma unroll
    for (int d = 0; d < 10; ++d) acc += e[d] * gate_w[d*24960 + idx];
    sgw[idx] = acc;
  }
  for (int idx = tid; idx < 128; idx += 256) {
    float acc = 0.f;
    #pragma unroll
    for (int d = 0; d < 10; ++d) acc += e[d] * gate_b[d*128 + idx];
    sgb[idx] = acc;
  }
  for (int idx = tid; idx < XCOLS; idx += 256) {
    sx[idx]  = X [(size_t)n*XCOLS + idx];
    sy1[idx] = Y1[(size_t)n*XCOLS + idx];
    sy2[idx] = Y2[(size_t)n*XCOLS + idx];
  }
  __syncthreads();

  const int b = tid >> 4, olo = tid & 15;   // 16 threads per batch element
  float acc[8];
  #pragma unroll
  for (int j = 0; j < 8; ++j) acc[j] = sgb[olo + j*16];
  for (int i = 0; i < CIN; ++i) {
    const float xv  = sx [b*CIN + i];
    const float y1v = sy1[b*CIN + i];
    const float y2v = sy2[b*CIN + i];
    const float* g0 = &sgw[i*128];
    const float* g1 = &sgw[(CIN + i)*128];
    const float* g2 = &sgw[(2*CIN + i)*128];
    #pragma unroll
    for (int j = 0; j < 8; ++j) {
      const int o = olo + j*16;
      acc[j] += xv*g0[o] + y1v*g1[o] + y2v*g2[o];
    }
  }
  #pragma unroll
  for (int j = 0; j < 8; ++j) {
    const int o = olo + j*16;
    const float sig = 1.f / (1.f + expf(-acc[j]));
    if (o < DOUTC) {   // z: build candidate input in place
      X[(size_t)n*XCOLS + b*CIN + 1 + o] =
          sig * H[((size_t)n*BB + b)*DOUTC + o];
    } else {           // r: save for state update
      Rbuf[((size_t)n*BB + b)*DOUTC + (o - DOUTC)] = sig;
    }
  }
}

// ---------------------------------------------------------------------------
// Candidate + state update: hc = tanh(...); H = r*H + (1-r)*hc; states[t]=H
// dynamic LDS: (12480 + 64 + 3*1040) floats = 62656 B
// ---------------------------------------------------------------------------
__global__ __launch_bounds__(256) void k_cand(const float* __restrict__ E,
                                              const float* __restrict__ update_w,
                                              const float* __restrict__ update_b,
                                              const float* __restrict__ X2,
                                              const float* __restrict__ Y1,
                                              const float* __restrict__ Y2,
                                              const float* __restrict__ Rbuf,
                                              float* __restrict__ H,
                                              float* __restrict__ states, int t) {
  extern __shared__ float sm[];
  float* suw = sm;               // 3*65*64 = 12480
  float* sub = suw + 12480;      // 64
  float* sx  = sub + 64;         // 1040
  float* sy1 = sx  + 1040;
  float* sy2 = sy1 + 1040;
  const int n = blockIdx.x, tid = threadIdx.x;
  float e[10];
  #pragma unroll
  for (int d = 0; d < 10; ++d) e[d] = E[n*10 + d];

  for (int idx = tid; idx < 12480; idx += 256) {
    float acc = 0.f;
    #pragma unroll
    for (int d = 0; d < 10; ++d) acc += e[d] * update_w[d*12480 + idx];
    suw[idx] = acc;
  }
  for (int idx = tid; idx < 64; idx += 256) {
    float acc = 0.f;
    #pragma unroll
    for (int d = 0; d < 10; ++d) acc += e[d] * update_b[d*64 + idx];
    sub[idx] = acc;
  }
  for (int idx = tid; idx < XCOLS; idx += 256) {
    sx[idx]  = X2[(size_t)n*XCOLS + idx];
    sy1[idx] = Y1[(size_t)n*XCOLS + idx];
    sy2[idx] = Y2[(size_t)n*XCOLS + idx];
  }
  __syncthreads();

  const int b = tid >> 4, olo = tid & 15;
  float acc[4];
  #pragma unroll
  for (int j = 0; j < 4; ++j) acc[j] = sub[olo + j*16];
  for (int i = 0; i < CIN; ++i) {
    const float xv  = sx [b*CIN + i];
    const float y1v = sy1[b*CIN + i];
    const float y2v = sy2[b*CIN + i];
    const float* u0 = &suw[i*64];
    const float* u1 = &suw[(CIN + i)*64];
    const float* u2 = &suw[(2*CIN + i)*64];
    #pragma unroll
    for (int j = 0; j < 4; ++j) {
      const int o = olo + j*16;
      acc[j] += xv*u0[o] + y1v*u1[o] + y2v*u2[o];
    }
  }
  #pragma unroll
  for (int j = 0; j < 4; ++j) {
    const int o = olo + j*16;
    const float hc = tanhf(acc[j]);
    const size_t hidx = ((size_t)n*BB + b)*DOUTC + o;
    const float r  = Rbuf[hidx];
    const float hn = r*H[hidx] + (1.f - r)*hc;
    H[hidx] = hn;
    states[(((size_t)t*NB + n)*BB + b)*DOUTC + o] = hn;
  }
}

// ---------------------------------------------------------------------------
// Fused transformer, output at t = T-1 only. One wave per token (b,n).
// K,V for all t and q at t=11 built in LDS; seq len 12 attention + out proj.
// ---------------------------------------------------------------------------
__global__ __launch_bounds__(256) void k_attn(const float* __restrict__ states,
                                              const float* __restrict__ win,
                                              const float* __restrict__ bin,
                                              const float* __restrict__ wout,
                                              const float* __restrict__ bout,
                                              float* __restrict__ O1) {
  __shared__ float sm[8][1712];
  const int wave = threadIdx.x >> 5, l = threadIdx.x & 31;
  const int tok = blockIdx.x*8 + wave;
  const int b = tok >> 11, n = tok & 2047;
  float* Ks = sm[wave];        // 12*64
  float* Vs = Ks + 768;        // 12*64
  float* qs = Vs + 768;        // 64
  float* ss = qs + 64;         // 64 scratch (token vec, then attn out)
  float* sc = ss + 64;         // 48 scores

  for (int t = 0; t < TT; ++t) {
    for (int c = l; c < 64; c += 32) {
      const int j = c >> 1;
      const float div = expf((float)(2*j) * (-0.14391156955f)); // -ln(1e4)/64
      const float ang = (float)t * div;
      const float pe  = (c & 1) ? cosf(ang) : sinf(ang);
      ss[c] = states[(((size_t)t*NB + n)*BB + b)*64 + c] + pe;
    }
    __syncthreads();
    for (int o = l; o < 64; o += 32) {
      float accK = bin[64 + o], accV = bin[128 + o];
      const float* rk = win + (64 + o)*64;
      const float* rv = win + (128 + o)*64;
      for (int c = 0; c < 64; ++c) {
        const float sv = ss[c];
        accK += rk[c]*sv; accV += rv[c]*sv;
      }
      Ks[t*64 + o] = accK; Vs[t*64 + o] = accV;
      if (t == TT - 1) {
        float accQ = bin[o];
        const float* rq = win + o*64;
        for (int c = 0; c < 64; ++c) accQ += rq[c]*ss[c];
        qs[o] = accQ;
      }
    }
    __syncthreads();
  }
  for (int idx = l; idx < 48; idx += 32) {
    const int h = idx / 12, t = idx % 12;
    float s = 0.f;
    for (int j = 0; j < 16; ++j) s += qs[h*16 + j]*Ks[t*64 + h*16 + j];
    sc[idx] = s * 0.25f;   // / sqrt(hd=16)
  }
  __syncthreads();
  if (l < 4) {
    const int h = l;
    float mx = -1e30f;
    for (int t = 0; t < 12; ++t) mx = fmaxf(mx, sc[h*12 + t]);
    float sum = 0.f;
    for (int t = 0; t < 12; ++t) { const float e2 = expf(sc[h*12+t]-mx); sc[h*12+t] = e2; sum += e2; }
    const float inv = 1.f/sum;
    for (int t = 0; t < 12; ++t) sc[h*12 + t] *= inv;
  }
  __syncthreads();
  for (int c = l; c < 64; c += 32) {
    const int h = c >> 4;
    float acc = 0.f;
    for (int t = 0; t < 12; ++t) acc += sc[h*12 + t]*Vs[t*64 + c];
    ss[c] = acc;
  }
  __syncthreads();
  for (int o = l; o < 64; o += 32) {
    float acc = bout[o];
    const float* ro = wout + o*64;
    for (int c = 0; c < 64; ++c) acc += ro[c]*ss[c];
    O1[(size_t)tok*64 + o] = acc;
  }
}

// ---------------------------------------------------------------------------
// Head: out[b,oc,n] = conv_b[oc] + sum_h cw0*o1*W[oc,0,h] + H*(cw1*W[oc,1,h]+cw2*W[oc,2,h])
// ---------------------------------------------------------------------------
__global__ __launch_bounds__(256) void k_head(const float* __restrict__ O1,
                                              const float* __restrict__ H,
                                              const float* __restrict__ cw,
                                              const float* __restrict__ conv_w,
                                              const float* __restrict__ conv_b,
                                              float* __restrict__ out) {
  const int tid = blockIdx.x*256 + threadIdx.x;
  if (tid >= BB*NB) return;
  const int b = tid >> 11, n = tid & 2047;
  float o1[64], hf[64];
  for (int c = 0; c < 64; ++c) {
    o1[c] = O1[(size_t)tid*64 + c];
    hf[c] = H[((size_t)n*BB + b)*64 + c];
  }
  const float c0 = cw[0], c1 = cw[1], c2 = cw[2];
  for (int oc = 0; oc < 12; ++oc) {
    float acc = conv_b[oc];
    const float* w0 = conv_w + (oc*3 + 0)*64;
    const float* w1 = conv_w + (oc*3 + 1)*64;
    const float* w2 = conv_w + (oc*3 + 2)*64;
    for (int h = 0; h < 64; ++h)
      acc += c0*o1[h]*w0[h] + hf[h]*(c1*w1[h] + c2*w2[h]);
    out[((size_t)b*12 + oc)*NB + n] = acc;
  }
}

// ---------------------------------------------------------------------------
extern "C" void kernel_launch(void* const* d_in, const int* in_sizes, int n_in,
                              void* d_out, int out_size, void* d_ws, size_t ws_size,
                              hipStream_t stream) {
  const float* source   = (const float*)d_in[0];
  const float* emb      = (const float*)d_in[2];
  const float* gate_w   = (const float*)d_in[3];
  const float* gate_b   = (const float*)d_in[4];
  const float* update_w = (const float*)d_in[5];
  const float* update_b = (const float*)d_in[6];
  const float* attn_in_w  = (const float*)d_in[7];
  const float* attn_in_b  = (const float*)d_in[8];
  const float* attn_out_w = (const float*)d_in[9];
  const float* attn_out_b = (const float*)d_in[10];
  const float* channel_w  = (const float*)d_in[11];
  const float* conv_w     = (const float*)d_in[12];
  const float* conv_b     = (const float*)d_in[13];
  float* out = (float*)d_out;

  float* ws = (float*)d_ws;
  float* A      = ws + OFF_A;
  float* A2     = ws + OFF_A2;
  float* X      = ws + OFF_X;
  float* Y1     = ws + OFF_Y1;
  float* Y2     = ws + OFF_Y2;
  float* Rbuf   = ws + OFF_R;
  float* H      = ws + OFF_H;
  float* states = ws + OFF_ST;
  float* O1     = ws + OFF_O1;

  const int GATE_LDS = (24960 + 128 + 3*XCOLS) * 4;   // 112832 B
  const int CAND_LDS = (12480 + 64 + 3*XCOLS) * 4;    //  62656 B
  (void)hipFuncSetAttribute((const void*)k_gate,
      hipFuncAttributeMaxDynamicSharedMemorySize, GATE_LDS);
  (void)hipFuncSetAttribute((const void*)k_cand,
      hipFuncAttributeMaxDynamicSharedMemorySize, CAND_LDS);

  const dim3 gGemmA2(8, 128);   // 2048x2048 output, BM=256
  const dim3 gGemmY(8, 65);     // 2048x1040 output, BM=256

  // --- precompute graph supports
  k_softmax_adj<<<NB, 256, 0, stream>>>(emb, A);
  k_gemm_wmma<<<gGemmA2, 256, 0, stream>>>(A, A, A2, 2048, 2048, 2048, 1);
  k_zero<<<(NB*BB*DOUTC + 255)/256, 256, 0, stream>>>(H, NB*BB*DOUTC);

  // --- GRU over time (states == last_states, computed once)
  for (int t = 0; t < TT; ++t) {
    k_build_X<<<(NB*XCOLS + 255)/256, 256, 0, stream>>>(X, source, H, t);
    k_gemm_wmma<<<gGemmY, 256, 0, stream>>>(A,  X, Y1, 2048, XCOLS, XCOLS, 0);
    k_gemm_wmma<<<gGemmY, 256, 0, stream>>>(A2, X, Y2, 2048, XCOLS, XCOLS, 0);
    k_gate<<<NB, 256, GATE_LDS, stream>>>(emb, gate_w, gate_b, Y1, Y2, H, Rbuf, X);
    k_gemm_wmma<<<gGemmY, 256, 0, stream>>>(A,  X, Y1, 2048, XCOLS, XCOLS, 0);
    k_gemm_wmma<<<gGemmY, 256, 0, stream>>>(A2, X, Y2, 2048, XCOLS, XCOLS, 0);
    k_cand<<<NB, 256, CAND_LDS, stream>>>(emb, update_w, update_b, X, Y1, Y2,
                                          Rbuf, H, states, t);
  }

  // --- transformer (last timestep only) + head
  k_attn<<<(BB*NB)/8, 256, 0, stream>>>(states, attn_in_w, attn_in_b,
                                        attn_out_w, attn_out_b, O1);
  k_head<<<(BB*NB + 255)/256, 256, 0, stream>>>(O1, H, channel_w, conv_w, conv_b, out);
}